// LUTApplier_7086696038792
// MI455X (gfx1250) — compile-verified
//
#include <hip/hip_runtime.h>
#include <stdint.h>

// Trilinear 3D-LUT apply: image (4,3,1080,1920) f32, lut (4,3,33,33,33) f32.
// Per-(batch,channel) LUT slice (143.7 KB) is staged into LDS by 8 parallel
// Tensor Data Mover descriptors (one per wave); a zeroed guard region makes
// all 8 trilinear corners compile-time immediate offsets from one base, so the
// inner loop is 8 ds_load_b32 + 7 fma-lerps per pixel. Image/output move in
// b128 quanta; output uses non-temporal stores to keep the image (read by 3
// channel-blocks) and LUT resident in the 192 MB L2.

#define LUT_N     33
#define LUT_NN    1089                 // 33*33
#define LUT_NNN   35937                // 33^3 elements = 143,748 bytes
#define LUT_TOT   37120                // + zeroed guard (needs >= 35937+1123), 16B aligned
#define IMG_HW    (1080 * 1920)        // 2,073,600 (divisible by 4)
#define PPB       16384                // pixels per block tile (divisible by 4)
#define THREADS   256
#define TDM_CHUNK 4496                 // elements per wave's TDM descriptor (8 waves)

typedef unsigned int v4u __attribute__((ext_vector_type(4)));
typedef int          v8i __attribute__((ext_vector_type(8)));
typedef int          v4i __attribute__((ext_vector_type(4)));
typedef float        v4f __attribute__((ext_vector_type(4)));

__global__ __launch_bounds__(THREADS)
void lut3d_tdm_kernel(const float* __restrict__ image,
                      const float* __restrict__ lut,
                      float* __restrict__ out)
{
    __shared__ __align__(16) float lut_s[LUT_TOT];

    const int bc = blockIdx.x;          // 0..11 fast: same-tile blocks adjacent -> L2 image reuse
    const int b  = bc / 3;

    // ---- Stage lut[b][c]: 8 TDM descriptors, one issued per wave ----
    {
        const int wv  = (int)(threadIdx.x >> 5);                 // wave id 0..7
        int off = wv * TDM_CHUNK;
        int cnt = LUT_NNN - off; if (cnt > TDM_CHUNK) cnt = TDM_CHUNK;  // last wave: 4465

        uint64_t ga    = (uint64_t)(uintptr_t)(lut + (size_t)bc * LUT_NNN)
                       + (uint64_t)(unsigned)off * 4u;
        uint32_t lds_a = ((uint32_t)(uintptr_t)(void*)&lut_s[0]) + (uint32_t)off * 4u;

        // D# group 0: count=1 | lds_addr | global_addr[56:0] | type=2
        v4u g0;
        g0.x = 1u;
        g0.y = lds_a;
        g0.z = (uint32_t)ga;
        g0.w = (uint32_t)(ga >> 32) | 0x80000000u;

        // D# group 1: 1-row tensor of `cnt` f32 elements.
        v8i g1;
        g1[0] = 0x00020000;                              // data_size=2 (4 bytes)
        g1[1] = (int)((unsigned)cnt << 16);              // tensor_dim0[15:0]
        g1[2] = (int)(1u << 16);                         // tensor_dim1 = 1
        g1[3] = (int)((unsigned)cnt << 16);              // tile_dim0 = cnt
        g1[4] = 1;                                       // tile_dim1 = 1, tile_dim2 = 0
        g1[5] = cnt;                                     // tensor_dim0_stride
        g1[6] = (int)(((unsigned)cnt & 0xFFFFu) << 16);  // tensor_dim1_stride[15:0]
        g1[7] = 0;

        v4i zg = {0, 0, 0, 0};
#if defined(__clang_major__) && (__clang_major__ >= 23)
        v8i zg8 = {0, 0, 0, 0, 0, 0, 0, 0};
        __builtin_amdgcn_tensor_load_to_lds(g0, g1, zg, zg, zg8, 0);
#else
        __builtin_amdgcn_tensor_load_to_lds(g0, g1, zg, zg, 0);
#endif
    }

    // Zero the guard padding while the DMA is in flight (disjoint LDS range).
    for (int i = LUT_NNN + (int)threadIdx.x; i < LUT_TOT; i += THREADS)
        lut_s[i] = 0.0f;

    __builtin_amdgcn_s_wait_tensorcnt(0);
    __syncthreads();

    // ---- Vectorized per-pixel trilinear interpolation from LDS ----
    const float* __restrict__ im0 = image + (size_t)b * 3 * IMG_HW;
    const v4f* __restrict__ xs4 = (const v4f*)im0;               // x coord plane
    const v4f* __restrict__ ys4 = (const v4f*)(im0 + IMG_HW);    // y
    const v4f* __restrict__ zs4 = (const v4f*)(im0 + 2 * IMG_HW);// z
    v4f* __restrict__ out4 = (v4f*)(out + (size_t)bc * IMG_HW);

    auto interp = [&](float xv, float yv, float zv) -> float {
        const float s = (float)(LUT_N - 1);
        xv = fminf(fmaxf(xv * s, 0.0f), s);
        yv = fminf(fmaxf(yv * s, 0.0f), s);
        zv = fminf(fmaxf(zv * s, 0.0f), s);
        float xf = floorf(xv), yf = floorf(yv), zf = floorf(zv);
        float wx = xv - xf, wy = yv - yf, wz = zv - zf;
        int i000 = ((int)zf * LUT_N + (int)yf) * LUT_N + (int)xf;
        // All offsets are compile-time: single VGPR address + immediate ds offsets.
        float c000 = lut_s[i000];
        float c001 = lut_s[i000 + 1];
        float c010 = lut_s[i000 + LUT_N];
        float c011 = lut_s[i000 + LUT_N + 1];
        float c100 = lut_s[i000 + LUT_NN];
        float c101 = lut_s[i000 + LUT_NN + 1];
        float c110 = lut_s[i000 + LUT_NN + LUT_N];
        float c111 = lut_s[i000 + LUT_NN + LUT_N + 1];
        float c00 = fmaf(wx, c001 - c000, c000);
        float c01 = fmaf(wx, c011 - c010, c010);
        float c10 = fmaf(wx, c101 - c100, c100);
        float c11 = fmaf(wx, c111 - c110, c110);
        float c0  = fmaf(wy, c01 - c00, c00);
        float c1  = fmaf(wy, c11 - c10, c10);
        return fmaf(wz, c1 - c0, c0);
    };

    int qbeg = (blockIdx.y * PPB) >> 2;
    int qend = qbeg + (PPB >> 2);
    const int qmax = IMG_HW >> 2;
    if (qend > qmax) qend = qmax;

    for (int q = qbeg + (int)threadIdx.x; q < qend; q += THREADS) {
        v4f xv = xs4[q], yv = ys4[q], zv = zs4[q];
        v4f r;
        r.x = interp(xv.x, yv.x, zv.x);
        r.y = interp(xv.y, yv.y, zv.y);
        r.z = interp(xv.z, yv.z, zv.z);
        r.w = interp(xv.w, yv.w, zv.w);
        __builtin_nontemporal_store(r, &out4[q]);
    }
}

extern "C" void kernel_launch(void* const* d_in, const int* in_sizes, int n_in,
                              void* d_out, int out_size, void* d_ws, size_t ws_size,
                              hipStream_t stream) {
    const float* image = (const float*)d_in[0];   // (4,3,1080,1920) f32
    const float* lut   = (const float*)d_in[1];   // (4,3,33,33,33)  f32
    float* out = (float*)d_out;                   // (4,3,1080,1920) f32

    const int tiles = (IMG_HW + PPB - 1) / PPB;   // 127
    dim3 grid(12, tiles, 1);                      // (b,c) fast dim
    lut3d_tdm_kernel<<<grid, THREADS, 0, stream>>>(image, lut, out);
}